// Proto_24043226923075
// MI455X (gfx1250) — compile-verified
//
#include <hip/hip_runtime.h>

typedef __attribute__((ext_vector_type(2))) float v2f;
typedef __attribute__((ext_vector_type(8))) float v8f;

#define NUM_CLASSES 21
#define CPAD 32
#define DDIM 768
#define NS 32768
#define NQ 32768

// workspace layout (floats)
#define OFF_PROTO 0                      // [32][768] padded proto
#define OFF_SUM   (CPAD * DDIM)          // 24576: [21][768] class sums
#define OFF_CNT   (OFF_SUM + NUM_CLASSES * DDIM)   // 40704: [21] counts
#define OFF_PNORM (OFF_CNT + 32)         // 40736: [32] proto norms
#define WS_FLOATS (OFF_PNORM + 32)       // 40768

// ---------------------------------------------------------------- zero ws
__global__ __launch_bounds__(256) void zero_ws_kernel(float* __restrict__ ws) {
    int i = blockIdx.x * 256 + threadIdx.x;
    if (i < WS_FLOATS) ws[i] = 0.0f;
}

// ------------------------------------------- per-class masked sums + counts
// 128 blocks x 256 threads; each block owns 256 support tokens.
// LDS-accumulate [21][768] sums + [21] counts, flush once with global atomics.
__global__ __launch_bounds__(256) void proto_accum_kernel(
    const float* __restrict__ emb, const int* __restrict__ lab,
    const int* __restrict__ msk, float* __restrict__ ws)
{
    __shared__ float lsum[NUM_CLASSES * DDIM];   // 63 KB
    __shared__ float lcnt[NUM_CLASSES];
    const int tid = threadIdx.x;
    for (int i = tid; i < NUM_CLASSES * DDIM; i += 256) lsum[i] = 0.0f;
    if (tid < NUM_CLASSES) lcnt[tid] = 0.0f;
    __syncthreads();

    const int base = blockIdx.x * 256;
    // counts: one token per thread
    {
        const int idx = base + tid;
        if (msk[idx] != 0) atomicAdd(&lcnt[lab[idx]], 1.0f);
    }
    // sums: all 256 threads cooperate per token (coalesced row reads)
    for (int t = 0; t < 256; ++t) {
        const int idx = base + t;
        if (msk[idx] != 0) {
            const int c = lab[idx];
            const float* __restrict__ row = emb + (size_t)idx * DDIM;
            float* __restrict__ dst = lsum + c * DDIM;
            atomicAdd(&dst[tid      ], row[tid      ]);
            atomicAdd(&dst[tid + 256], row[tid + 256]);
            atomicAdd(&dst[tid + 512], row[tid + 512]);
        }
    }
    __syncthreads();
    float* __restrict__ gsum = ws + OFF_SUM;
    float* __restrict__ gcnt = ws + OFF_CNT;
    for (int i = tid; i < NUM_CLASSES * DDIM; i += 256) atomicAdd(&gsum[i], lsum[i]);
    if (tid < NUM_CLASSES) atomicAdd(&gcnt[tid], lcnt[tid]);
}

// ------------------------------------------------ proto = sum/cnt, + ||p||^2
// 32 blocks (one per padded class) x 256 threads.
__global__ __launch_bounds__(256) void proto_finalize_kernel(float* __restrict__ ws)
{
    const int c = blockIdx.x;           // 0..31
    const int tid = threadIdx.x;
    const float* __restrict__ gsum = ws + OFF_SUM;
    const float* __restrict__ gcnt = ws + OFF_CNT;
    float* __restrict__ proto = ws + OFF_PROTO;
    float* __restrict__ pnorm = ws + OFF_PNORM;

    const float cnt = (c < NUM_CLASSES) ? gcnt[c] : 0.0f;
    const float inv = (cnt > 0.0f) ? (1.0f / fmaxf(cnt, 1.0f)) : 0.0f;

    float acc = 0.0f;
    for (int d = tid; d < DDIM; d += 256) {
        const float v = (c < NUM_CLASSES) ? gsum[c * DDIM + d] * inv : 0.0f;
        proto[c * DDIM + d] = v;
        acc += v * v;
    }
    __shared__ float red[256];
    red[tid] = acc;
    __syncthreads();
    for (int s = 128; s > 0; s >>= 1) {
        if (tid < s) red[tid] += red[tid + s];
        __syncthreads();
    }
    if (tid == 0) pnorm[c] = red[0];
}

// -------------------------------------- logits + argmax via f32 WMMA 16x16x4
// 256 blocks x 256 threads (8 waves). Block: 128 queries x 32 (padded) classes.
// Wave: 16 queries x 32 classes = two 16x16 f32 accumulators, K=768.
#define QT_STRIDE 36   // LDS row stride: 16B-aligned, conflict-free for 16-lane row reads

__global__ __launch_bounds__(256) void dist_wmma_kernel(
    const float* __restrict__ Q, const float* __restrict__ ws, float* __restrict__ out)
{
    const float* __restrict__ proto = ws + OFF_PROTO;   // [32][768]
    const float* __restrict__ pnorm = ws + OFF_PNORM;   // [32]
    float* __restrict__ logits = out;                           // [NQ][21]
    float* __restrict__ pred   = out + (size_t)NQ * NUM_CLASSES;

    __shared__ float qtile[128 * QT_STRIDE];   // K-chunk of 32 per query row
    __shared__ float qnorm_s[8 * 16];

    const int tid  = threadIdx.x;
    const int wave = tid >> 5;
    const int lane = tid & 31;
    const int qblk = blockIdx.x * 128;
    const int q0   = qblk + wave * 16;

    const int m    = lane & 15;          // A row / B col / D column index
    const int koff = (lane >> 4) * 2;    // K sub-offset for A/B fragments

    v8f acc0 = {};   // classes 0..15
    v8f acc1 = {};   // classes 16..31
    float qn = 0.0f;
    const int nc0 = (lane >> 4) * 16;    // 2 lanes per row split the 32 norm cols

    for (int kc = 0; kc < DDIM; kc += 32) {
        // stage Q[qblk..+127][kc..+15] into LDS, fully coalesced float4 loads
        #pragma unroll
        for (int i = 0; i < 2; ++i) {
            const int idx = tid + i * 256;          // 0..511
            const int row = idx >> 2;
            const int c4  = (idx & 3) * 4;
            const float4 v = *(const float4*)(Q + (size_t)(qblk + row) * DDIM + kc + c4);
            float* dstp = &qtile[row * QT_STRIDE + c4];
            dstp[0] = v.x; dstp[1] = v.y; dstp[2] = v.z; dstp[3] = v.w;
        }
        __syncthreads();

        // partial ||q||^2 from the staged tile (saves a second 96MB pass over Q)
        {
            const float* r = &qtile[(wave * 16 + m) * QT_STRIDE + nc0];
            #pragma unroll
            for (int c = 0; c < 16; ++c) { const float x = r[c]; qn += x * x; }
        }

        // 8 WMMA k-steps of 4 over this chunk
        #pragma unroll
        for (int kk = 0; kk < 8; ++kk) {
            const int kl = kk * 4 + koff;
            v2f a;
            const float* ap = &qtile[(wave * 16 + m) * QT_STRIDE + kl];
            a.x = ap[0]; a.y = ap[1];
            const int kg = kc + kl;
            v2f b0, b1;
            const float* bp0 = proto + (size_t)m * DDIM + kg;
            b0.x = bp0[0]; b0.y = bp0[1];
            const float* bp1 = proto + (size_t)(16 + m) * DDIM + kg;
            b1.x = bp1[0]; b1.y = bp1[1];
            acc0 = __builtin_amdgcn_wmma_f32_16x16x4_f32(false, a, false, b0,
                                                         (short)0, acc0, false, false);
            acc1 = __builtin_amdgcn_wmma_f32_16x16x4_f32(false, a, false, b1,
                                                         (short)0, acc1, false, false);
        }
        __syncthreads();
    }

    // combine the two half-row norm partials; publish per-row norms via LDS
    qn += __shfl_xor(qn, 16, 32);
    if (lane < 16) qnorm_s[wave * 16 + lane] = qn;
    __syncthreads();

    const float pn0 = pnorm[m];
    const float pn1 = pnorm[16 + m];

    #pragma unroll
    for (int r = 0; r < 8; ++r) {
        const int   mr  = (lane < 16) ? r : (r + 8);
        const float qnr = qnorm_s[wave * 16 + mr];
        const float l0  = 2.0f * acc0[r] - qnr - pn0;   // class m
        const float l1  = 2.0f * acc1[r] - qnr - pn1;   // class 16+m (pad >=21 dropped)
        const int qrow  = q0 + mr;

        logits[(size_t)qrow * NUM_CLASSES + m] = l0;
        if (m < NUM_CLASSES - 16) logits[(size_t)qrow * NUM_CLASSES + 16 + m] = l1;

        // per-row argmax over 21 classes (lowest-index tie-break, matches jnp.argmax)
        float bv = l0; int bi = m;
        if (m < NUM_CLASSES - 16 && l1 > bv) { bv = l1; bi = 16 + m; }
        #pragma unroll
        for (int off = 1; off < 16; off <<= 1) {
            const float ov = __shfl_xor(bv, off, 32);
            const int   oi = __shfl_xor(bi, off, 32);
            if (ov > bv || (ov == bv && oi < bi)) { bv = ov; bi = oi; }
        }
        if (m == 0) pred[qrow] = (float)bi;
    }
}

// ---------------------------------------------------------------- launcher
extern "C" void kernel_launch(void* const* d_in, const int* in_sizes, int n_in,
                              void* d_out, int out_size, void* d_ws, size_t ws_size,
                              hipStream_t stream) {
    const float* semb  = (const float*)d_in[0];   // [256,128,768] f32
    const float* qemb  = (const float*)d_in[1];   // [256,128,768] f32
    const int*   slab  = (const int*)d_in[2];     // [256,128] i32
    const int*   smask = (const int*)d_in[3];     // [256,128] i32
    float* ws  = (float*)d_ws;
    float* out = (float*)d_out;

    zero_ws_kernel<<<(WS_FLOATS + 255) / 256, 256, 0, stream>>>(ws);
    proto_accum_kernel<<<NS / 256, 256, 0, stream>>>(semb, slab, smask, ws);
    proto_finalize_kernel<<<CPAD, 256, 0, stream>>>(ws);
    dist_wmma_kernel<<<NQ / 128, 256, 0, stream>>>(qemb, ws, out);
}